// VGAE_encoder_47863115547311
// MI455X (gfx1250) — compile-verified
//
#include <hip/hip_runtime.h>

#define NV 50000
#define EV 800000
#define FD 128

typedef __attribute__((ext_vector_type(2))) float v2f;
typedef __attribute__((ext_vector_type(8))) float v8f;

// Guaranteed single-instruction fp32 scatter-add (no CAS fallback).
// No-return form -> tracked by STOREcnt; S_ENDPGM's implicit wait-idle and
// stream-ordered kernel launches make results visible to later kernels.
__device__ __forceinline__ void gadd_f32(float* p, float v) {
  asm volatile("global_atomic_add_f32 %0, %1, off"
               :
               : "v"(p), "v"(v)
               : "memory");
}

// ---------------- init kernels ----------------
__global__ void k_zero(float* __restrict__ p, int n) {
  int i = blockIdx.x * blockDim.x + threadIdx.x;
  if (i < n) p[i] = 0.0f;
}

__global__ void k_fill(float* __restrict__ p, float v, int n) {
  int i = blockIdx.x * blockDim.x + threadIdx.x;
  if (i < n) p[i] = v;
}

// out[:, :64] (mu) = b_mu ; out[:, 64:] (lv) = b_lv  (flat: mu block then lv block)
__global__ void k_bias_init(float* __restrict__ out,
                            const float* __restrict__ bmu,
                            const float* __restrict__ blv) {
  int i = blockIdx.x * blockDim.x + threadIdx.x;
  if (i >= NV * FD) return;
  int node = i >> 7, f = i & 127;
  if (f < 64) out[(size_t)node * 64 + f] = bmu[f];
  else        out[(size_t)NV * 64 + (size_t)node * 64 + (f - 64)] = blv[f - 64];
}

// ---------------- degree / norm ----------------
__global__ void k_degree(const long long* __restrict__ ei, float* __restrict__ deg) {
  int e = blockIdx.x * blockDim.x + threadIdx.x;
  if (e < EV) gadd_f32(&deg[(int)ei[EV + e]], 1.0f);
}

__global__ void k_rsqrt(float* __restrict__ deg) {
  int i = blockIdx.x * blockDim.x + threadIdx.x;
  if (i < NV) {
    float d = deg[i];
    deg[i] = (d > 0.0f) ? rsqrtf(d) : 0.0f;
  }
}

// ---------------- WMMA f32 GEMM: C[tile] = A(MxK) * B(KxN) ----------------
// One wave per 16x16 output tile. K stepped by 4 via V_WMMA_F32_16X16X4_F32.
// A layout: lanes 0-15 -> rows M, K={k,k+1}; lanes 16-31 -> K={k+2,k+3}.
// B layout mirrored on columns. D layout per ISA 16x16 f32 C/D table.
__global__ void k_gemm_wmma(const float* __restrict__ A, int lda,
                            const float* __restrict__ B, int ldb,
                            float* __restrict__ C, int ldc, int coff,
                            int tilesN, int totalTiles) {
  int wave = (blockIdx.x * blockDim.x + threadIdx.x) >> 5;
  if (wave >= totalTiles) return;
  int lane = threadIdx.x & 31;
  int half = lane >> 4;        // 0 or 1
  int l15  = lane & 15;
  int tm = wave / tilesN;
  int tn = wave % tilesN;

  const float* Ap = A + (size_t)(tm * 16 + l15) * lda + 2 * half;
  const float* Bp = B + (size_t)(2 * half) * ldb + tn * 16 + l15;

  v8f acc = {};
  #pragma unroll 4
  for (int k = 0; k < FD; k += 4) {
    v2f a = *(const v2f*)(Ap + k);            // K=k+2*half, k+2*half+1 (contiguous)
    v2f b;
    b.x = Bp[(size_t)k * ldb];                // K=k+2*half
    b.y = Bp[(size_t)k * ldb + ldb];          // K=k+2*half+1
    acc = __builtin_amdgcn_wmma_f32_16x16x4_f32(
        false, a, false, b, (short)0, acc, false, false);
  }

  float* Cp = C + (size_t)(tm * 16 + 8 * half) * ldc + tn * 16 + coff + l15;
  #pragma unroll
  for (int r = 0; r < 8; ++r) Cp[(size_t)r * ldc] = acc[r];
}

// ---------------- edge scatter: agg[dst] += h[src]*norm (conv1, 128 feat) ----
__global__ void k_scatter1(const float* __restrict__ h,
                           const long long* __restrict__ ei,
                           const float* __restrict__ dis,
                           float* __restrict__ agg) {
  int wave = (blockIdx.x * blockDim.x + threadIdx.x) >> 5;
  if (wave >= EV) return;
  int lane = threadIdx.x & 31;
  int s = (int)ei[wave];
  int d = (int)ei[EV + wave];
  float w = dis[s] * dis[d];
  float4 v = *((const float4*)(h + (size_t)s * FD) + lane);
  float* op = agg + (size_t)d * FD + lane * 4;
  gadd_f32(op + 0, v.x * w);
  gadd_f32(op + 1, v.y * w);
  gadd_f32(op + 2, v.z * w);
  gadd_f32(op + 3, v.w * w);
}

// self-loop contribution (ordered after scatter -> non-atomic)
__global__ void k_self1(const float* __restrict__ h, const float* __restrict__ dis,
                        float* __restrict__ agg) {
  int i = blockIdx.x * blockDim.x + threadIdx.x;
  if (i >= NV * FD) return;
  int node = i >> 7;
  float w = dis[node]; w *= w;
  agg[i] += h[i] * w;
}

__global__ void k_relu_bias(float* __restrict__ agg, const float* __restrict__ b1) {
  int i = blockIdx.x * blockDim.x + threadIdx.x;
  if (i >= NV * FD) return;
  float v = agg[i] + b1[i & 127];
  agg[i] = v > 0.0f ? v : 0.0f;
}

// ---------------- edge scatter conv2: features 0-63 -> mu, 64-127 -> logvar ---
__global__ void k_scatter2(const float* __restrict__ hcat,
                           const long long* __restrict__ ei,
                           const float* __restrict__ dis,
                           float* __restrict__ out) {
  int wave = (blockIdx.x * blockDim.x + threadIdx.x) >> 5;
  if (wave >= EV) return;
  int lane = threadIdx.x & 31;
  int s = (int)ei[wave];
  int d = (int)ei[EV + wave];
  float w = dis[s] * dis[d];
  float4 v = *((const float4*)(hcat + (size_t)s * FD) + lane);
  int f = lane * 4;  // [f, f+3] lies entirely in one half (f<64 for lanes<16)
  float* base = (f < 64)
      ? (out + (size_t)d * 64 + f)
      : (out + (size_t)NV * 64 + (size_t)d * 64 + (f - 64));
  gadd_f32(base + 0, v.x * w);
  gadd_f32(base + 1, v.y * w);
  gadd_f32(base + 2, v.z * w);
  gadd_f32(base + 3, v.w * w);
}

__global__ void k_self2(const float* __restrict__ hcat, const float* __restrict__ dis,
                        float* __restrict__ out) {
  int i = blockIdx.x * blockDim.x + threadIdx.x;
  if (i >= NV * FD) return;
  int node = i >> 7, f = i & 127;
  float w = dis[node]; w *= w;
  float v = hcat[i] * w;
  if (f < 64) out[(size_t)node * 64 + f] += v;
  else        out[(size_t)NV * 64 + (size_t)node * 64 + (f - 64)] += v;
}

// ---------------- host launch ----------------
extern "C" void kernel_launch(void* const* d_in, const int* in_sizes, int n_in,
                              void* d_out, int out_size, void* d_ws, size_t ws_size,
                              hipStream_t stream) {
  const float*     x   = (const float*)d_in[0];
  const long long* ei  = (const long long*)d_in[1];   // int64 [2, E]
  const float*     W1  = (const float*)d_in[2];
  const float*     b1  = (const float*)d_in[3];
  const float*     Wmu = (const float*)d_in[4];
  const float*     bmu = (const float*)d_in[5];
  const float*     Wlv = (const float*)d_in[6];
  const float*     blv = (const float*)d_in[7];
  float* out = (float*)d_out;

  float* ws = (float*)d_ws;
  float* dis  = ws;                       // 50000 f32 (deg in-place -> dis)
  float* h1   = ws + 65536;               // 6.4M f32 (reused as hcat for conv2)
  float* agg  = ws + 65536 + NV * FD;     // 6.4M f32 (conv1 aggregate -> h)
  (void)in_sizes; (void)n_in; (void)out_size; (void)ws_size;

  const int T = 256;
  const int NF_BLK = (NV * FD + T - 1) / T;       // 25000
  const int N_BLK  = (NV + T - 1) / T;            // 196
  const int E_BLK  = (EV + T - 1) / T;            // 3125
  const int EW_BLK = (EV + (T / 32) - 1) / (T / 32); // wave-per-edge: 100000

  // init
  k_zero<<<NF_BLK, T, 0, stream>>>(agg, NV * FD);
  k_fill<<<N_BLK, T, 0, stream>>>(dis, 1.0f, NV);        // self-loop counts
  k_bias_init<<<NF_BLK, T, 0, stream>>>(out, bmu, blv);

  // degree -> deg^-1/2
  k_degree<<<E_BLK, T, 0, stream>>>(ei, dis);
  k_rsqrt<<<N_BLK, T, 0, stream>>>(dis);

  // conv1: h1 = x @ W1  (3125x8 = 25000 tiles, 8 waves/block)
  k_gemm_wmma<<<3125, T, 0, stream>>>(x, FD, W1, FD, h1, FD, 0, 8, 25000);
  k_scatter1<<<EW_BLK, T, 0, stream>>>(h1, ei, dis, agg);
  k_self1<<<NF_BLK, T, 0, stream>>>(h1, dis, agg);
  k_relu_bias<<<NF_BLK, T, 0, stream>>>(agg, b1);        // agg now holds h

  // conv2: hcat = h @ [Wmu | Wlv]  (each 3125x4 = 12500 tiles)
  k_gemm_wmma<<<1563, T, 0, stream>>>(agg, FD, Wmu, 64, h1, FD, 0,  4, 12500);
  k_gemm_wmma<<<1563, T, 0, stream>>>(agg, FD, Wlv, 64, h1, FD, 64, 4, 12500);
  k_scatter2<<<EW_BLK, T, 0, stream>>>(h1, ei, dis, out);
  k_self2<<<NF_BLK, T, 0, stream>>>(h1, dis, out);
}